// LightGCN_4681514352907
// MI455X (gfx1250) — compile-verified
//
#include <hip/hip_runtime.h>

typedef __attribute__((ext_vector_type(2))) float v2f;
typedef __attribute__((ext_vector_type(8))) float v8f;

#define DIM 64
#define LAYERS 4

static inline int cdiv_i(long long a, long long b) { return (int)((a + b - 1) / b); }
static inline int capgrid(long long n, int cap) { int g = cdiv_i(n, 256); return g > cap ? cap : (g < 1 ? 1 : g); }

// ---------------- elementwise helpers ----------------

__global__ void lg_zero4(float4* __restrict__ p, long long n4) {
  long long i = (long long)blockIdx.x * blockDim.x + threadIdx.x;
  long long st = (long long)gridDim.x * blockDim.x;
  float4 z = make_float4(0.f, 0.f, 0.f, 0.f);
  for (; i < n4; i += st) p[i] = z;
}

__global__ void lg_degree(const int* __restrict__ dst, float* __restrict__ deg, int E) {
  int i = blockIdx.x * blockDim.x + threadIdx.x;
  int st = gridDim.x * blockDim.x;
  for (; i < E; i += st) atomicAdd(&deg[dst[i]], 1.0f);
}

__global__ void lg_rsqrt(float* __restrict__ d, int n) {
  int i = blockIdx.x * blockDim.x + threadIdx.x;
  int st = gridDim.x * blockDim.x;
  for (; i < n; i += st) {
    float v = d[i];
    d[i] = (v > 0.f) ? rsqrtf(v) : 0.f;
  }
}

__global__ void lg_weight(const int* __restrict__ src, const int* __restrict__ dst,
                          const float* __restrict__ dis, float* __restrict__ w, int E) {
  int i = blockIdx.x * blockDim.x + threadIdx.x;
  int st = gridDim.x * blockDim.x;
  for (; i < E; i += st) w[i] = dis[src[i]] * dis[dst[i]];
}

// out_final = out_orig = x = src (3-way broadcast copy, vectorized)
__global__ void lg_copy3(const float4* __restrict__ s, float4* __restrict__ a,
                         float4* __restrict__ b, float4* __restrict__ c, int n4) {
  int i = blockIdx.x * blockDim.x + threadIdx.x;
  int st = gridDim.x * blockDim.x;
  for (; i < n4; i += st) {
    float4 v = s[i];
    a[i] = v; b[i] = v; c[i] = v;
  }
}

__global__ void lg_accum(float4* __restrict__ d, const float4* __restrict__ s, int n4) {
  int i = blockIdx.x * blockDim.x + threadIdx.x;
  int st = gridDim.x * blockDim.x;
  for (; i < n4; i += st) {
    float4 v = s[i];
    float4 u = d[i];
    u.x += v.x; u.y += v.y; u.z += v.z; u.w += v.w;
    d[i] = u;
  }
}

// ---------------- scatter: one wave32 per edge, float2 per lane ----------------
// Per wave: uniform (scalar) loads of src/dst/w, one coalesced 256B row gather
// from L2-resident x, 64 f32 atomic adds into x_new. Prefetch next edge's row.
__global__ void lg_scatter(const int* __restrict__ src, const int* __restrict__ dst,
                           const float* __restrict__ w, const float* __restrict__ x,
                           float* __restrict__ xn, int E) {
  const int lane = threadIdx.x & 31;
  int wv = blockIdx.x * (blockDim.x >> 5) + (threadIdx.x >> 5);
  const int nw = gridDim.x * (blockDim.x >> 5);
  for (int e = wv; e < E; e += nw) {
    int s = src[e];
    int d = dst[e];
    float we = w[e];
    int e2 = e + nw;
    if (e2 < E) {
      int s2 = src[e2];
      __builtin_prefetch(x + (size_t)s2 * DIM + lane * 2, 0, 0); // global_prefetch_b8
    }
    const v2f* xs = (const v2f*)(x + (size_t)s * DIM);
    v2f v = xs[lane];
    float* xd = xn + (size_t)d * DIM + lane * 2;
    atomicAdd(xd, we * v.x);
    atomicAdd(xd + 1, we * v.y);
  }
}

// ---------------- finalize: exact scale-by-s via chained V_WMMA_F32_16X16X4_F32 ----
// s*T (16x16 tile) = sum_j  A_j(16x4 = T[:,4j:4j+4]) x B_j(4x16 = rows 4j..4j+3 of s*I16)
// Each D element gets exactly one nonzero f32 product (RNE) + exact zeros -> bit-equal
// to elementwise multiply. One wave per 16x16 tile; tile predicate is wave-uniform so
// EXEC is all-ones inside the WMMA region.
__global__ void lg_finalize_wmma(float* __restrict__ base, int nrowtiles, float scale) {
  const int lane = threadIdx.x & 31;
  int tile = blockIdx.x * (blockDim.x >> 5) + (threadIdx.x >> 5);
  int ntiles = nrowtiles * (DIM / 16);
  if (tile >= ntiles) return; // uniform per wave
  int rt = tile >> 2;         // row tile (16 rows)
  int ct = tile & 3;          // column tile (16 of 64 features)
  float* t0 = base + (size_t)rt * 16 * DIM + (size_t)ct * 16;
  int half = lane >> 4;       // A/B: lanes 0-15 hold K={0,1}, lanes 16-31 K={2,3}
  int m = lane & 15;
  v8f acc = {0.f, 0.f, 0.f, 0.f, 0.f, 0.f, 0.f, 0.f};
#pragma unroll
  for (int j = 0; j < 4; ++j) {
    // A_j: lane -> row m, columns 4j + 2*half + {0,1}
    const float* ap = t0 + (size_t)m * DIM + 4 * j + 2 * half;
    v2f a;
    a.x = ap[0];
    a.y = ap[1];
    // B_j[k][n] = (n == 4j+k) ? scale : 0 ; VGPR0 holds K=2*half, VGPR1 holds K=2*half+1
    v2f b;
    b.x = (m == 4 * j + 2 * half)     ? scale : 0.0f;
    b.y = (m == 4 * j + 2 * half + 1) ? scale : 0.0f;
    acc = __builtin_amdgcn_wmma_f32_16x16x4_f32(
        /*neg_a=*/false, a, /*neg_b=*/false, b,
        /*c_mod=*/(short)0, acc, /*reuse_a=*/false, /*reuse_b=*/false);
  }
  // D layout: VGPR r -> lanes 0-15: (M=r, N=lane) ; lanes 16-31: (M=8+r, N=lane-16)
  float* op = t0 + (size_t)(half * 8) * DIM + m;
#pragma unroll
  for (int r = 0; r < 8; ++r) op[(size_t)r * DIM] = acc[r];
}

// ---------------- driver ----------------

extern "C" void kernel_launch(void* const* d_in, const int* in_sizes, int n_in,
                              void* d_out, int out_size, void* d_ws, size_t ws_size,
                              hipStream_t stream) {
  (void)n_in; (void)out_size; (void)ws_size;

  const float* emb_users = (const float*)d_in[0];
  const float* emb_items = (const float*)d_in[1];
  const int*   edge      = (const int*)d_in[2];

  const int NU = in_sizes[0] / DIM;     // 100000
  const int NI = in_sizes[1] / DIM;     // 50000
  const int N  = NU + NI;               // 150000
  const int E  = in_sizes[2] / 2;       // 4,000,000
  const int* src = edge;
  const int* dst = edge + E;

  // workspace layout (floats): dis[N] | w[E] | xA[N*DIM] | xB[N*DIM]
  float* ws  = (float*)d_ws;
  float* dis = ws;
  float* w   = dis + N;
  float* xA  = w + (size_t)E;
  float* xB  = xA + (size_t)N * DIM;

  // output layout: users_final | users_orig | items_final | items_orig
  float* out        = (float*)d_out;
  float* usersFinal = out;
  float* usersOrig  = out + (size_t)NU * DIM;
  float* itemsFinal = out + (size_t)2 * NU * DIM;
  float* itemsOrig  = itemsFinal + (size_t)NI * DIM;

  const int B = 256;
  const float scale = 1.0f / (float)((LAYERS + 1) * (LAYERS + 1));

  // 1) degree -> dis
  lg_zero4<<<capgrid(N / 4, 2048), B, 0, stream>>>((float4*)dis, N / 4);
  lg_degree<<<capgrid(E, 4096), B, 0, stream>>>(dst, dis, E);
  lg_rsqrt<<<capgrid(N, 2048), B, 0, stream>>>(dis, N);

  // 2) per-edge symmetric-normalized weight
  lg_weight<<<capgrid(E, 4096), B, 0, stream>>>(src, dst, dis, w, E);

  // 3) init: x = emb, embs_sum (in d_out final slots) = emb, orig copies = emb
  lg_copy3<<<capgrid((long long)NU * DIM / 4, 4096), B, 0, stream>>>(
      (const float4*)emb_users, (float4*)xA, (float4*)usersFinal, (float4*)usersOrig,
      NU * DIM / 4);
  lg_copy3<<<capgrid((long long)NI * DIM / 4, 4096), B, 0, stream>>>(
      (const float4*)emb_items, (float4*)(xA + (size_t)NU * DIM), (float4*)itemsFinal,
      (float4*)itemsOrig, NI * DIM / 4);

  // 4) propagation layers
  float* xin = xA;
  float* xout = xB;
  for (int l = 0; l < LAYERS; ++l) {
    lg_zero4<<<capgrid((long long)N * DIM / 4, 4096), B, 0, stream>>>(
        (float4*)xout, (long long)N * DIM / 4);
    lg_scatter<<<4096, B, 0, stream>>>(src, dst, w, xin, xout, E);
    lg_accum<<<capgrid((long long)NU * DIM / 4, 4096), B, 0, stream>>>(
        (float4*)usersFinal, (const float4*)xout, NU * DIM / 4);
    lg_accum<<<capgrid((long long)NI * DIM / 4, 4096), B, 0, stream>>>(
        (float4*)itemsFinal, (const float4*)(xout + (size_t)NU * DIM), NI * DIM / 4);
    float* t = xin; xin = xout; xout = t;
  }

  // 5) exact scale by 1/25 via WMMA (rows are multiples of 16: 100000, 50000)
  {
    int nrt_u = NU / 16, nrt_i = NI / 16;
    int ntiles_u = nrt_u * (DIM / 16), ntiles_i = nrt_i * (DIM / 16);
    int wpb = B / 32;
    lg_finalize_wmma<<<cdiv_i(ntiles_u, wpb), B, 0, stream>>>(usersFinal, nrt_u, scale);
    lg_finalize_wmma<<<cdiv_i(ntiles_i, wpb), B, 0, stream>>>(itemsFinal, nrt_i, scale);
  }
}